// GIN_31121333027434
// MI455X (gfx1250) — compile-verified
//
#include <hip/hip_runtime.h>

#define N_NODES 100000
#define N_EDGES 1600000
#define FDIM    128

typedef float v2f __attribute__((ext_vector_type(2)));
typedef float v8f __attribute__((ext_vector_type(8)));

// ---------------------------------------------------------------------------
// Guaranteed-native f32 atomic adds (no CAS-loop fallback): inline CDNA5 asm,
// no-return form -> tracked by STOREcnt, drained by implicit wait at s_endpgm.
// Four consecutive floats via the instruction offset field.
// ---------------------------------------------------------------------------
__device__ __forceinline__ void atomic_add_f32_x4(float* p, float4 v) {
    asm volatile(
        "global_atomic_add_f32 %0, %1, off\n\t"
        "global_atomic_add_f32 %0, %2, off offset:4\n\t"
        "global_atomic_add_f32 %0, %3, off offset:8\n\t"
        "global_atomic_add_f32 %0, %4, off offset:12"
        :
        : "v"(p), "v"(v.x), "v"(v.y), "v"(v.z), "v"(v.w)
        : "memory");
}

// ---------------------------------------------------------------------------
// Scatter: agg[dst[e]] += x[src[e]]   (32 lanes per edge, float4 per lane)
// ---------------------------------------------------------------------------
__global__ void gin_scatter(const float* __restrict__ X,
                            const long long* __restrict__ src,
                            const long long* __restrict__ dst,
                            float* __restrict__ AGG) {
    unsigned gtid = blockIdx.x * blockDim.x + threadIdx.x;
    unsigned e = gtid >> 5;            // edge index (wave-uniform)
    if (e >= N_EDGES) return;
    unsigned c = (gtid & 31u) * 4u;    // feature offset (0..124)

    long long s = src[e];
    long long d = dst[e];

    const float4 v = *(const float4*)(X + (size_t)s * FDIM + c);  // b128 load
    atomic_add_f32_x4(AGG + (size_t)d * FDIM + c, v);
}

// ---------------------------------------------------------------------------
// WMMA GEMM: Y[16 rows x NOUT] = (X + AGG)[16 x 128] @ W[128 x NOUT] + B
// One wave per 16-row tile; NT = ceil(NOUT/16) accumulators held in VGPRs.
// Uses V_WMMA_F32_16X16X4_F32 (exact fp32 math, matches reference dtype).
// In-place (Y == X) is safe: each wave exclusively owns its 16 rows.
// ---------------------------------------------------------------------------
template <int NT, int NOUT>
__global__ void gin_gemm(const float* __restrict__ X,
                         const float* __restrict__ AGG,
                         const float* __restrict__ W,
                         const float* __restrict__ B,
                         float* __restrict__ Y,
                         int n_row_tiles) {
    int wave = (int)((blockIdx.x * blockDim.x + threadIdx.x) >> 5);
    if (wave >= n_row_tiles) return;   // uniform per wave -> EXEC stays full

    int lane = threadIdx.x & 31;
    int half = lane >> 4;              // 0: lanes 0-15, 1: lanes 16-31
    int l16  = lane & 15;
    int kb   = half * 2;               // A/B fragment K base (ISA 16x4 layout)
    int row  = wave * 16 + l16;        // always < 100000 (6250*16 exact)

    const float* xrow = X   + (size_t)row * FDIM;
    const float* grow = AGG + (size_t)row * FDIM;

    v8f acc[NT];
#pragma unroll
    for (int t = 0; t < NT; ++t) acc[t] = (v8f){};

    for (int k0 = 0; k0 < FDIM; k0 += 4) {
        // A fragment: lane (half,l16) holds A[M=l16][K=k0+kb], [K=k0+kb+1]
        v2f a;
        a.x = xrow[k0 + kb]     + grow[k0 + kb];
        a.y = xrow[k0 + kb + 1] + grow[k0 + kb + 1];

#pragma unroll
        for (int t = 0; t < NT; ++t) {
            int col = t * 16 + l16;
            // arithmetic masking (select, not branch) keeps EXEC all-1s
            int cc = (NOUT % 16 == 0) ? col : (col < NOUT ? col : 0);
            float m = (NOUT % 16 == 0) ? 1.0f : (col < NOUT ? 1.0f : 0.0f);
            v2f b;
            b.x = W[(size_t)(k0 + kb)     * NOUT + cc] * m;
            b.y = W[(size_t)(k0 + kb + 1) * NOUT + cc] * m;
            acc[t] = __builtin_amdgcn_wmma_f32_16x16x4_f32(
                false, a, false, b, (short)0, acc[t], false, false);
        }
    }

    // Bias + store.  C/D layout: vgpr r -> M = r + 8*half, lane -> N = l16.
#pragma unroll
    for (int t = 0; t < NT; ++t) {
        int col = t * 16 + l16;
        if (NOUT % 16 != 0 && col >= NOUT) continue;  // store-only divergence OK
        float bv = B[col];
#pragma unroll
        for (int r = 0; r < 8; ++r) {
            int m = r + half * 8;
            Y[(size_t)(wave * 16 + m) * NOUT + col] = acc[t][r] + bv;
        }
    }
}

// ---------------------------------------------------------------------------
// Launch: 5 GIN layers.  Scratch: agg (51.2MB) + one ping buffer (51.2MB).
// Hidden layers run the GEMM in place on the ping buffer.
// ---------------------------------------------------------------------------
extern "C" void kernel_launch(void* const* d_in, const int* in_sizes, int n_in,
                              void* d_out, int out_size, void* d_ws, size_t ws_size,
                              hipStream_t stream) {
    const float* h     = (const float*)d_in[0];   // 100000 x 128
    const float* Ws    = (const float*)d_in[1];   // 4 x 128 x 128
    const float* bs    = (const float*)d_in[2];   // 4 x 128
    const float* W_out = (const float*)d_in[3];   // 128 x 40
    const float* b_out = (const float*)d_in[4];   // 40
    const long long* ei = (const long long*)d_in[5]; // 2 x N_EDGES (int64)
    const long long* src = ei;
    const long long* dst = ei + N_EDGES;

    float* agg = (float*)d_ws;                    // N_NODES*128 floats
    float* buf = agg + (size_t)N_NODES * FDIM;    // N_NODES*128 floats

    const size_t agg_bytes = (size_t)N_NODES * FDIM * sizeof(float);

    const int scat_threads = 256;
    const int scat_blocks  = (int)(((long long)N_EDGES * 32 + scat_threads - 1) / scat_threads);

    const int row_tiles = N_NODES / 16;           // 6250 exactly
    const int gemm_threads = 256;                 // 8 waves per block
    const int gemm_blocks  = (row_tiles + 7) / 8;

    // Layer 0: X = h (immutable input) -> buf
    hipMemsetAsync(agg, 0, agg_bytes, stream);
    gin_scatter<<<scat_blocks, scat_threads, 0, stream>>>(h, src, dst, agg);
    gin_gemm<8, 128><<<gemm_blocks, gemm_threads, 0, stream>>>(
        h, agg, Ws + 0 * FDIM * FDIM, bs + 0 * FDIM, buf, row_tiles);

    // Layers 1..3: in-place on buf
    for (int l = 1; l < 4; ++l) {
        hipMemsetAsync(agg, 0, agg_bytes, stream);
        gin_scatter<<<scat_blocks, scat_threads, 0, stream>>>(buf, src, dst, agg);
        gin_gemm<8, 128><<<gemm_blocks, gemm_threads, 0, stream>>>(
            buf, agg, Ws + (size_t)l * FDIM * FDIM, bs + (size_t)l * FDIM,
            buf, row_tiles);
    }

    // Layer 4: buf -> d_out (128 -> 40, 3 col tiles with masking)
    hipMemsetAsync(agg, 0, agg_bytes, stream);
    gin_scatter<<<scat_blocks, scat_threads, 0, stream>>>(buf, src, dst, agg);
    gin_gemm<3, 40><<<gemm_blocks, gemm_threads, 0, stream>>>(
        buf, agg, W_out, b_out, (float*)d_out, row_tiles);
}